// MultitaskLossA_79740362817929
// MI455X (gfx1250) — compile-verified
//
#include <hip/hip_runtime.h>
#include <hip/hip_bf16.h>
#include <math.h>

typedef __attribute__((ext_vector_type(2))) float v2f;
typedef __attribute__((ext_vector_type(4))) float v4f;
typedef __attribute__((ext_vector_type(8))) float v8f;

#define ALPHA_C 1.0f

__device__ __forceinline__ float bce_term(float x, float y) {
    // numerically stable BCE-with-logits, identical formula to reference
    return fmaxf(x, 0.0f) - x * y + log1pf(expf(-fabsf(x)));
}

// ---------------------------------------------------------------------------
// Wave(32)-wide exact f32 sum via V_WMMA_F32_16X16X4_F32 with a ones B-matrix.
// A is 16x4 f32: lanes 0-15 supply {K0 = x, K1 = 0}; lanes 16-31 supply
// {K2 = x, K3 = 0}.  With B = ones, D[m,n] = x[m] + x[m+16] for all n.
// D layout: lanes 0-15 hold rows M=0..7 in VGPRs 0..7, lanes 16-31 rows 8..15,
// so summing the 8 D registers per lane and one xor-16 add yields the full
// 32-lane sum in every lane.  All arithmetic stays in f32 (multiplies by 1.0),
// so this is an exact tree reduction.
// ---------------------------------------------------------------------------
__device__ __forceinline__ float wave_sum32(float x) {
    v2f a; a[0] = x;    a[1] = 0.0f;
    v2f b; b[0] = 1.0f; b[1] = 1.0f;
    v8f c = {};
    v8f d = __builtin_amdgcn_wmma_f32_16x16x4_f32(
        /*neg_a=*/false, a, /*neg_b=*/false, b,
        /*c_mod=*/(short)0, c, /*reuse_a=*/false, /*reuse_b=*/false);
    float s = ((d[0] + d[1]) + (d[2] + d[3])) + ((d[4] + d[5]) + (d[6] + d[7]));
    s += __shfl_xor(s, 16, 32);
    return s;
}

// Block-level sum (blockDim.x == 256 -> 8 waves). Result valid in thread 0.
__device__ __forceinline__ float block_sum256(float x) {
    __shared__ float lds[8];
    float w = wave_sum32(x);
    int lane = threadIdx.x & 31;
    int wid  = threadIdx.x >> 5;
    if (lane == 0) lds[wid] = w;
    __syncthreads();
    float t = 0.0f;
    if (threadIdx.x == 0) {
        int nw = (int)(blockDim.x >> 5);
        for (int i = 0; i < nw; ++i) t += lds[i];
    }
    return t;
}

__global__ __launch_bounds__(256) void MultitaskLossA_main(
        const float* __restrict__ pb,   // y_pred_binary  [B]
        const float* __restrict__ pt,   // y_pred_type    [B,T]
        const float* __restrict__ ps,   // y_pred_source  [B,32]
        const float* __restrict__ tb,   // y_true_binary  [B]
        const float* __restrict__ tt,   // y_true_type    [B,T]
        const int*   __restrict__ ts,   // y_true_source  [B]
        float* __restrict__ partials,
        int Bn, int Tn)
{
    const int gtid = blockIdx.x * blockDim.x + threadIdx.x;
    const int nthr = gridDim.x * blockDim.x;
    const float invB  = 1.0f / (float)Bn;
    const float invBT = 1.0f / ((float)Bn * (float)Tn);

    float acc = 0.0f;

    // ---- Phase A: binary BCE, mean over B -------------------------------
    {
        const v4f* x4 = (const v4f*)pb;
        const v4f* y4 = (const v4f*)tb;
        const int n4 = Bn >> 2;
        for (int i = gtid; i < n4; i += nthr) {
            v4f x = __builtin_nontemporal_load(&x4[i]);
            v4f y = __builtin_nontemporal_load(&y4[i]);
            float s = bce_term(x[0], y[0]) + bce_term(x[1], y[1])
                    + bce_term(x[2], y[2]) + bce_term(x[3], y[3]);
            acc += s * invB;
        }
        for (int i = (n4 << 2) + gtid; i < Bn; i += nthr)  // tail (none for B=2^21)
            acc += bce_term(pb[i], tb[i]) * invB;
    }

    // ---- Phase B: type BCE, mean over B*T -------------------------------
    {
        const v4f* x4 = (const v4f*)pt;
        const v4f* y4 = (const v4f*)tt;
        const long long n  = (long long)Bn * (long long)Tn;
        const int n4 = (int)(n >> 2);
        for (int i = gtid; i < n4; i += nthr) {
            v4f x = __builtin_nontemporal_load(&x4[i]);
            v4f y = __builtin_nontemporal_load(&y4[i]);
            float s = bce_term(x[0], y[0]) + bce_term(x[1], y[1])
                    + bce_term(x[2], y[2]) + bce_term(x[3], y[3]);
            acc += s * invBT;
        }
        for (long long i = ((long long)n4 << 2) + gtid; i < n; i += nthr)
            acc += bce_term(pt[i], tt[i]) * invBT;
    }

    // ---- Phase C: 32-way cross-entropy + consistency, wave per row ------
    // One wave per row: lane c holds logit c -> every load is one fully
    // coalesced 128B line.  The argmax butterfly also yields the row max,
    // so only two butterflies (+1 broadcast) are needed per row.
    {
        const int wave   = gtid >> 5;
        const int lane   = gtid & 31;
        const int nwaves = nthr >> 5;
        for (int r = wave; r < Bn; r += nwaves) {
            float x = __builtin_nontemporal_load(ps + ((size_t)r << 5) + lane);

            // fused max + argmax, first-index tie-break (matches jnp.argmax)
            float v  = x;
            int   id = lane;
            for (int off = 16; off > 0; off >>= 1) {
                float ov = __shfl_xor(v, off, 32);
                int   oi = __shfl_xor(id, off, 32);
                if (ov > v || (ov == v && oi < id)) { v = ov; id = oi; }
            }
            // v == row max in every lane now

            // sum of exp(x - max)
            float s = expf(x - v);
            for (int off = 16; off > 0; off >>= 1)
                s += __shfl_xor(s, off, 32);

            int   label = ts[r];
            float xl    = __shfl(x, label, 32);

            if (lane == 0) {
                float ce = (v + logf(s)) - xl;         // -log_softmax[label]
                acc += ce * invB;
                // round(sigmoid(x)) > 0  <=>  x > 0  (0.5 rounds half-even to 0)
                bool bcls = pb[r] > 0.0f;
                bool tcls = pt[(size_t)r * (size_t)Tn] > 0.0f;
                bool scls = id > 0;
                float mism = (float)(bcls != tcls) + (float)(bcls != scls);
                acc += (ALPHA_C * invB) * mism;
            }
        }
    }

    // ---- deterministic block reduction (WMMA-based wave sums) -----------
    float bsum = block_sum256(acc);
    if (threadIdx.x == 0) partials[blockIdx.x] = bsum;
}

__global__ __launch_bounds__(256) void MultitaskLossA_final(
        const float* __restrict__ partials, float* __restrict__ out, int n)
{
    float acc = 0.0f;
    for (int i = threadIdx.x; i < n; i += blockDim.x)
        acc += partials[i];
    float total = block_sum256(acc);
    if (threadIdx.x == 0) out[0] = total;
}

extern "C" void kernel_launch(void* const* d_in, const int* in_sizes, int n_in,
                              void* d_out, int out_size, void* d_ws, size_t ws_size,
                              hipStream_t stream) {
    const float* pb = (const float*)d_in[0];
    const float* pt = (const float*)d_in[1];
    const float* ps = (const float*)d_in[2];
    const float* tb = (const float*)d_in[3];
    const float* tt = (const float*)d_in[4];
    const int*   ts = (const int*)d_in[5];

    const int Bn = in_sizes[0];
    const int Tn = (Bn > 0) ? (in_sizes[1] / Bn) : 1;   // T = 8

    float* partials = (float*)d_ws;

    const int NB = 2048;   // 2048 blocks * 8 waves = 16384 waves in flight
    const int NT = 256;

    MultitaskLossA_main<<<NB, NT, 0, stream>>>(pb, pt, ps, tb, tt, ts,
                                               partials, Bn, Tn);
    MultitaskLossA_final<<<1, NT, 0, stream>>>(partials, (float*)d_out, NB);
}